// GBST_26061861552188
// MI455X (gfx1250) — compile-verified
//
#include <hip/hip_runtime.h>
#include <hip/hip_bf16.h>
#include <math.h>

typedef __attribute__((ext_vector_type(2))) float v2f;
typedef __attribute__((ext_vector_type(8))) float v8f;

#define BATCH  4
#define NSEQ   4096
#define DIM    512
#define LPAD   4104          // lcm(1,2,3,4)=12 padding of 4096
#define KCONV  4
#define TOUT   1024          // NSEQ / DS
#define TJ     512           // attention j-tile

// ---------------------------------------------------------------------------
// 1) token gather + depthwise conv (pad K-1 at end) : Y[b*n, d]
// ---------------------------------------------------------------------------
__global__ void k_gather_conv(const int* __restrict__ x,
                              const float* __restrict__ emb,
                              const float* __restrict__ cw,   // [DIM,K]
                              const float* __restrict__ cb,   // [DIM]
                              float* __restrict__ Y)
{
    const int row = blockIdx.x;            // 0..B*N-1
    const int i   = row & (NSEQ - 1);
    int tok[KCONV];
#pragma unroll
    for (int k = 0; k < KCONV; ++k)
        tok[k] = (i + k < NSEQ) ? x[row + k] : -1;

#pragma unroll
    for (int rep = 0; rep < 2; ++rep) {
        const int d = threadIdx.x + rep * 256;
        const float4 w = ((const float4*)cw)[d];     // cw[d][0..3]
        float acc = cb[d];
        if (tok[0] >= 0) acc += emb[(size_t)tok[0] * DIM + d] * w.x;
        if (tok[1] >= 0) acc += emb[(size_t)tok[1] * DIM + d] * w.y;
        if (tok[2] >= 0) acc += emb[(size_t)tok[2] * DIM + d] * w.z;
        if (tok[3] >= 0) acc += emb[(size_t)tok[3] * DIM + d] * w.w;
        Y[(size_t)row * DIM + d] = acc;
    }
}

// ---------------------------------------------------------------------------
// 2) zero the pad rows of H (rows NSEQ..LPAD-1 per batch)
// ---------------------------------------------------------------------------
__global__ void k_zero_pad(float* __restrict__ H)
{
    const int idx = blockIdx.x * blockDim.x + threadIdx.x;   // 0..B*8*DIM-1
    const int d   = idx & (DIM - 1);
    const int t   = idx >> 9;            // 0..31
    const int b   = t >> 3;
    const int pr  = t & 7;
    H[((size_t)b * LPAD + NSEQ + pr) * DIM + d] = 0.0f;
}

// ---------------------------------------------------------------------------
// 3) H = Y @ proj_w^T + proj_b  via V_WMMA_F32_16X16X4_F32.
//    Each wave: one 16(M)x64(N) tile, 4 accumulators, K=512 in steps of 4.
//    A 16x4 f32 lane layout: lane l holds M=l&15, K pair starting at 2*(l>>4).
//    B 4x16 layout symmetric: lane l holds N=l&15, same K pair.
//    B[k, o] = W[o, k]  (W = proj_w [out,in] row-major)  -> contiguous pair.
// ---------------------------------------------------------------------------
__global__ void k_gemm_wmma(const float* __restrict__ Y,
                            const float* __restrict__ W,
                            const float* __restrict__ bias,
                            float* __restrict__ H)
{
    const int wave = blockIdx.x * (blockDim.x >> 5) + (threadIdx.x >> 5);
    const int lane = threadIdx.x & 31;
    const int mtile = wave >> 3;         // 0..1023
    const int ngrp  = wave & 7;          // 0..7
    const int r0 = mtile * 16;
    const int n0 = ngrp * 64;
    const int m  = lane & 15;
    const int kg = lane >> 4;            // 0 or 1

    const float* yrow = Y + (size_t)(r0 + m) * DIM;
    const float* w0 = W + (size_t)(n0 +  0 + m) * DIM;
    const float* w1 = W + (size_t)(n0 + 16 + m) * DIM;
    const float* w2 = W + (size_t)(n0 + 32 + m) * DIM;
    const float* w3 = W + (size_t)(n0 + 48 + m) * DIM;

    v8f acc0 = {}, acc1 = {}, acc2 = {}, acc3 = {};

#pragma unroll 4
    for (int k0 = 0; k0 < DIM; k0 += 4) {
        const int ko = k0 + 2 * kg;
        v2f a  = *(const v2f*)(yrow + ko);
        v2f b0 = *(const v2f*)(w0 + ko);
        v2f b1 = *(const v2f*)(w1 + ko);
        v2f b2 = *(const v2f*)(w2 + ko);
        v2f b3 = *(const v2f*)(w3 + ko);
        acc0 = __builtin_amdgcn_wmma_f32_16x16x4_f32(false, a, false, b0, (short)0, acc0, false, false);
        acc1 = __builtin_amdgcn_wmma_f32_16x16x4_f32(false, a, false, b1, (short)0, acc1, false, false);
        acc2 = __builtin_amdgcn_wmma_f32_16x16x4_f32(false, a, false, b2, (short)0, acc2, false, false);
        acc3 = __builtin_amdgcn_wmma_f32_16x16x4_f32(false, a, false, b3, (short)0, acc3, false, false);
    }

    // Epilogue: C/D layout -> VGPR v: M = v + 8*(lane>>4), N = lane&15.
    const int bidx = r0 >> 12;           // batch (4096 rows per batch, 16 | 4096)
    const int i0   = r0 - bidx * NSEQ;
    float* hb = H + ((size_t)bidx * LPAD + i0) * DIM;
    v8f accs[4] = {acc0, acc1, acc2, acc3};
#pragma unroll
    for (int t = 0; t < 4; ++t) {
        const int col = n0 + t * 16 + m;
        const float bv = bias[col];
#pragma unroll
        for (int v = 0; v < 8; ++v) {
            const int rowi = v + 8 * kg;
            hb[(size_t)rowi * DIM + col] = accs[t][v] + bv;
        }
    }
}

// ---------------------------------------------------------------------------
// 4) s[b,i] = dot(H[b,i,:], score_w)   (one wave per row, shfl reduction)
// ---------------------------------------------------------------------------
__global__ void k_score_proj(const float* __restrict__ H,
                             const float* __restrict__ sw,
                             float* __restrict__ S)
{
    const int wid  = threadIdx.x >> 5;
    const int lane = threadIdx.x & 31;
    const int row  = blockIdx.x * 8 + wid;    // 0..B*LPAD-1
    if (row >= BATCH * LPAD) return;

    const float4* hp = (const float4*)(H + (size_t)row * DIM);
    const float4* wp = (const float4*)sw;
    float sum = 0.0f;
#pragma unroll
    for (int q = 0; q < 4; ++q) {
        const float4 hv = hp[lane + 32 * q];
        const float4 wv = wp[lane + 32 * q];
        sum += hv.x * wv.x + hv.y * wv.y + hv.z * wv.z + hv.w * wv.w;
    }
#pragma unroll
    for (int off = 16; off >= 1; off >>= 1)
        sum += __shfl_xor(sum, off, 32);
    if (lane == 0) S[row] = sum;
}

// ---------------------------------------------------------------------------
// 5) per-position block scores (block-mean of s) + softmax over nb=4
// ---------------------------------------------------------------------------
__global__ void k_block_scores(const float* __restrict__ S,
                               const float* __restrict__ sb,
                               float4* __restrict__ SC0)
{
    const int idx = blockIdx.x * blockDim.x + threadIdx.x;
    if (idx >= BATCH * LPAD) return;
    const int b = idx / LPAD;
    const int i = idx - b * LPAD;
    const float* s = S + (size_t)b * LPAD;
    const float bias = sb[0];

    const float v0 = s[i] + bias;
    const int i2 = i & ~1;
    const float v1 = 0.5f * (s[i2] + s[i2 + 1]) + bias;
    const int i3 = (i / 3) * 3;
    const float v2 = (s[i3] + s[i3 + 1] + s[i3 + 2]) * (1.0f / 3.0f) + bias;
    const int i4 = i & ~3;
    const float v3 = 0.25f * (s[i4] + s[i4 + 1] + s[i4 + 2] + s[i4 + 3]) + bias;

    const float mx = fmaxf(fmaxf(v0, v1), fmaxf(v2, v3));
    const float e0 = __expf(v0 - mx), e1 = __expf(v1 - mx);
    const float e2 = __expf(v2 - mx), e3 = __expf(v3 - mx);
    const float inv = 1.0f / (e0 + e1 + e2 + e3);
    SC0[idx] = make_float4(e0 * inv, e1 * inv, e2 * inv, e3 * inv);
}

// ---------------------------------------------------------------------------
// 6) consensus attention (flash-style online softmax, d=4, V==K==scores)
// ---------------------------------------------------------------------------
__global__ void k_attn(const float4* __restrict__ SC0,
                       float4* __restrict__ FSC)
{
    __shared__ float4 tile[TJ];
    const int nblk = (LPAD + 255) / 256;          // 17
    const int b = blockIdx.x / nblk;
    const int i = (blockIdx.x % nblk) * 256 + threadIdx.x;
    const float4* sc = SC0 + (size_t)b * LPAD;

    float4 q = make_float4(0.f, 0.f, 0.f, 0.f);
    const bool valid = (i < LPAD);
    if (valid) q = sc[i];

    float m = -INFINITY, l = 0.0f;
    float ax = 0.f, ay = 0.f, az = 0.f, aw = 0.f;

    for (int j0 = 0; j0 < LPAD; j0 += TJ) {
        const int cnt = (LPAD - j0 < TJ) ? (LPAD - j0) : TJ;
        __syncthreads();
        for (int jj = threadIdx.x; jj < cnt; jj += blockDim.x)
            tile[jj] = sc[j0 + jj];
        __syncthreads();
        for (int jj = 0; jj < cnt; ++jj) {
            const float4 kv = tile[jj];
            const float v  = q.x * kv.x + q.y * kv.y + q.z * kv.z + q.w * kv.w;
            const float nm = fmaxf(m, v);
            const float c  = __expf(m - nm);   // exp(-inf)=0 handles first iter
            const float e  = __expf(v - nm);
            l  = l * c + e;
            ax = ax * c + e * kv.x;
            ay = ay * c + e * kv.y;
            az = az * c + e * kv.z;
            aw = aw * c + e * kv.w;
            m  = nm;
        }
    }
    if (valid) {
        const float inv = 1.0f / l;
        FSC[(size_t)b * LPAD + i] = make_float4(ax * inv, ay * inv, az * inv, aw * inv);
    }
}

// ---------------------------------------------------------------------------
// 7) fuse block representations (block means recomputed from H on the fly)
//    + downsample-by-4 mean. One block per (b, t) output row.
// ---------------------------------------------------------------------------
__global__ void k_fuse(const float* __restrict__ H,
                       const float4* __restrict__ FSC,
                       float* __restrict__ out)
{
    const int blk = blockIdx.x;               // 0..B*TOUT-1
    const int b = blk >> 10;
    const int t = blk & 1023;
    const int base = 4 * t;

    __shared__ float4 fs[4];
    if (threadIdx.x < 4) fs[threadIdx.x] = FSC[(size_t)b * LPAD + base + threadIdx.x];
    __syncthreads();

    const float* hrow = H + (size_t)b * LPAD * DIM;
#pragma unroll
    for (int rep = 0; rep < 2; ++rep) {
        const int d = threadIdx.x + rep * 256;
        float hv[8];                           // h[base-2 .. base+5]
#pragma unroll
        for (int jj = 0; jj < 8; ++jj) {
            const int j = base - 2 + jj;       // base+5 <= 4097 < LPAD
            hv[jj] = (j >= 0) ? hrow[(size_t)j * DIM + d] : 0.0f;
        }
        const float m4 = 0.25f * (hv[2] + hv[3] + hv[4] + hv[5]);
        float acc = 0.0f;
#pragma unroll
        for (int ii = 0; ii < 4; ++ii) {
            const float4 w = fs[ii];
            const float m1 = hv[2 + ii];
            const int p2 = ii & ~1;
            const float m2 = 0.5f * (hv[2 + p2] + hv[3 + p2]);
            const int off3 = ((base + ii) / 3) * 3 - base;   // in [-2,3]
            const float m3 = (hv[2 + off3] + hv[3 + off3] + hv[4 + off3]) * (1.0f / 3.0f);
            acc += w.x * m1 + w.y * m2 + w.z * m3 + w.w * m4;
        }
        out[((size_t)(b * TOUT + t)) * DIM + d] = 0.25f * acc;
    }
}

// ---------------------------------------------------------------------------
extern "C" void kernel_launch(void* const* d_in, const int* in_sizes, int n_in,
                              void* d_out, int out_size, void* d_ws, size_t ws_size,
                              hipStream_t stream)
{
    const int*   x       = (const int*)  d_in[0];
    const float* emb     = (const float*)d_in[1];
    const float* conv_w  = (const float*)d_in[2];
    const float* conv_b  = (const float*)d_in[3];
    const float* proj_w  = (const float*)d_in[4];
    const float* proj_b  = (const float*)d_in[5];
    const float* score_w = (const float*)d_in[6];
    const float* score_b = (const float*)d_in[7];
    float* out = (float*)d_out;

    char* ws = (char*)d_ws;
    const size_t Y_BYTES  = (size_t)BATCH * NSEQ * DIM * sizeof(float);   // 33.5 MB
    const size_t H_BYTES  = (size_t)BATCH * LPAD * DIM * sizeof(float);   // 33.6 MB
    const size_t S_BYTES  = (size_t)BATCH * LPAD * sizeof(float);
    const size_t SC_BYTES = (size_t)BATCH * LPAD * 4 * sizeof(float);

    float*  Y   = (float*) (ws);
    float*  H   = (float*) (ws + Y_BYTES);
    float*  S   = (float*) (ws + Y_BYTES + H_BYTES);
    float4* SC0 = (float4*)(ws + Y_BYTES + H_BYTES + S_BYTES);
    float4* FSC = (float4*)(ws + Y_BYTES + H_BYTES + S_BYTES + SC_BYTES);

    // 1) gather + depthwise conv
    k_gather_conv<<<BATCH * NSEQ, 256, 0, stream>>>(x, emb, conv_w, conv_b, Y);
    // 2) zero pad rows of H
    k_zero_pad<<<(BATCH * 8 * DIM) / 256, 256, 0, stream>>>(H);
    // 3) WMMA fp32 GEMM: 1024 M-tiles x 8 N-groups = 8192 waves, 4 waves/block
    k_gemm_wmma<<<2048, 128, 0, stream>>>(Y, proj_w, proj_b, H);
    // 4) score projection (8 waves/block)
    k_score_proj<<<(BATCH * LPAD + 7) / 8, 256, 0, stream>>>(H, score_w, S);
    // 5) block-mean scores + softmax over 4
    k_block_scores<<<(BATCH * LPAD + 255) / 256, 256, 0, stream>>>(S, score_b, SC0);
    // 6) consensus attention over L positions
    k_attn<<<BATCH * ((LPAD + 255) / 256), 256, 0, stream>>>(SC0, FSC);
    // 7) fuse + downsample
    k_fuse<<<BATCH * TOUT, 256, 0, stream>>>(H, FSC, out);
}